// DrugTargetGAT_38594576122354
// MI455X (gfx1250) — compile-verified
//
#include <hip/hip_runtime.h>
#include <math.h>

#define NEG_SLOPE 0.2f

typedef __attribute__((ext_vector_type(2))) float v2f;
typedef __attribute__((ext_vector_type(8))) float v8f;

// ---------- helpers ----------

__device__ __forceinline__ float lrelu(float x) { return x > 0.f ? x : NEG_SLOPE * x; }

// float atomic max via ordered-int trick (valid for all finite floats, init to -inf)
__device__ __forceinline__ void atomicMaxF(float* addr, float val) {
  if (val >= 0.f) atomicMax((int*)addr, __float_as_int(val));
  else            atomicMin((unsigned int*)addr, __float_as_uint(val));
}

__device__ __forceinline__ void atomicAddF(float* addr, float val) {
  unsafeAtomicAdd(addr, val);   // lowers to global_atomic_add_f32 on gfx1250
}

// ---------- generic fill (vectorized, b128 stores) ----------

__global__ void fill_kernel(float* __restrict__ p, float v, int n) {
  int i = blockIdx.x * blockDim.x + threadIdx.x;
  int i4 = i * 4;
  if (i4 + 3 < n) {
    float4 v4 = make_float4(v, v, v, v);
    ((float4*)p)[i] = v4;
  } else {
    for (int k = i4; k < n; ++k) p[k] = v;
  }
}

// ---------- degree / mean edge weight per dst ----------

__global__ void deg_kernel(const int* __restrict__ dst, const float* __restrict__ ew,
                           float* __restrict__ deg, float* __restrict__ wsum, int E) {
  int e = blockIdx.x * blockDim.x + threadIdx.x;
  if (e >= E) return;
  int d = dst[e];
  atomicAddF(&deg[d], 1.f);
  atomicAddF(&wsum[d], ew[e]);
}

__global__ void loopea_kernel(const float* __restrict__ deg, const float* __restrict__ wsum,
                              float* __restrict__ lea, int N) {
  int n = blockIdx.x * blockDim.x + threadIdx.x;
  if (n < N) lea[n] = wsum[n] / fmaxf(deg[n], 1.f);
}

// ---------- small constants: S1[h] = sum_c We1[h*32+c]*attE1[h*32+c]; c2 = We2*attE2 ----------

__global__ void const_kernel(const float* __restrict__ We1, const float* __restrict__ attE1,
                             const float* __restrict__ We2, const float* __restrict__ attE2,
                             float* __restrict__ cst) {
  int h = threadIdx.x;
  if (h < 4) {
    float s = 0.f;
    for (int c = 0; c < 32; ++c) s += We1[h * 32 + c] * attE1[h * 32 + c];
    cst[h] = s;
  }
  if (h == 4) cst[4] = We2[0] * attE2[0];
  if (h == 5) cst[5] = 0.f;
  if (h == 6) cst[6] = 0.f;
  if (h == 7) cst[7] = 0.f;
}

// ---------- layer-1 GEMM: h1[N,128] = x[N,2] @ W1[2,128] via V_WMMA_F32_16X16X4_F32 ----------
// A is 16x4 (K=2..3 zero-padded), B is 4x16 (rows 2..3 zero). One wave -> 16x128 tile.

__global__ void __launch_bounds__(128)
gemm1_wmma_kernel(const float* __restrict__ x, const float* __restrict__ W1,
                  float* __restrict__ h1, int N) {
  int wave = threadIdx.x >> 5;
  int lane = threadIdx.x & 31;
  int tile = blockIdx.x * 4 + wave;          // 16-row tile
  int row0 = tile * 16;
  if (row0 >= N) return;                     // uniform within a wave -> EXEC stays all-1s
  bool full = (row0 + 16 <= N);

  // A layout (16x4 f32): lanes 0-15: VGPR0=K0, VGPR1=K1 (M=lane); lanes 16-31: K2,K3 (zero pad)
  int r = row0 + (lane & 15);
  v2f a;
  if (lane < 16 && r < N) { a.x = x[2 * r]; a.y = x[2 * r + 1]; }
  else                    { a.x = 0.f;      a.y = 0.f; }

  // per-lane output base: row = row0 + (lane>=16 ? 8 : 0) + rr, col = j*16 + (lane&15)
  int myrow = row0 + (lane >> 4) * 8;
  int nsub  = lane & 15;

#pragma unroll
  for (int j = 0; j < 8; ++j) {
    // B layout (4x16 f32): lanes 0-15: VGPR0=K0 row, VGPR1=K1 row (N=lane); lanes 16-31: K2,K3 (zero)
    v2f b;
    if (lane < 16) { int col = j * 16 + lane; b.x = W1[col]; b.y = W1[128 + col]; }
    else           { b.x = 0.f; b.y = 0.f; }

    v8f c = {};
    c = __builtin_amdgcn_wmma_f32_16x16x4_f32(false, a, false, b, (short)0, c, false, false);

    // D layout: VGPR rr, lanes 0-15 -> (M=rr, N=lane); lanes 16-31 -> (M=8+rr, N=lane-16)
    float* outp = h1 + (size_t)myrow * 128 + j * 16 + nsub;
    if (full) {
      // fast path: 8 stores, row stride 512B folds into the instruction offset
#pragma unroll
      for (int rr = 0; rr < 8; ++rr) outp[rr * 128] = c[rr];
    } else {
#pragma unroll
      for (int rr = 0; rr < 8; ++rr) {
        if (myrow + rr < N) outp[rr * 128] = c[rr];
      }
    }
  }
}

// ---------- per-node, per-head attention scalars (b128 loads) ----------

__global__ void node_att1_kernel(const float* __restrict__ h1,
                                 const float* __restrict__ attS1, const float* __restrict__ attD1,
                                 float* __restrict__ aS, float* __restrict__ aD, int N) {
  int i = blockIdx.x * blockDim.x + threadIdx.x;
  if (i >= N * 4) return;
  int n = i >> 2, h = i & 3;
  const float4* hp = (const float4*)(h1 + (size_t)n * 128 + h * 32);
  const float4* sp = (const float4*)(attS1 + h * 32);
  const float4* dp = (const float4*)(attD1 + h * 32);
  float as = 0.f, ad = 0.f;
#pragma unroll
  for (int c = 0; c < 8; ++c) {
    float4 v = hp[c], s4 = sp[c], d4 = dp[c];
    as += v.x * s4.x + v.y * s4.y + v.z * s4.z + v.w * s4.w;
    ad += v.x * d4.x + v.y * d4.y + v.z * d4.z + v.w * d4.w;
  }
  aS[i] = as; aD[i] = ad;
}

// ---------- layer-1 edge softmax (3 passes over E real edges + N self-loops) ----------

__device__ __forceinline__ void edge_ids1(int i, int E, const int* src, const int* dst,
                                          const float* ew, const float* lea,
                                          int& s, int& d, float& w) {
  if (i < E) { s = src[i]; d = dst[i]; w = ew[i]; }
  else       { s = d = i - E; w = lea[s]; }
}

__global__ void edge_max1_kernel(const int* __restrict__ src, const int* __restrict__ dst,
                                 const float* __restrict__ ew, const float* __restrict__ lea,
                                 const float* __restrict__ aS, const float* __restrict__ aD,
                                 const float* __restrict__ cst, float* __restrict__ m1,
                                 int E, int N) {
  int i = blockIdx.x * blockDim.x + threadIdx.x;
  if (i >= E + N) return;
  int s, d; float w;
  edge_ids1(i, E, src, dst, ew, lea, s, d, w);
  float4 as4 = ((const float4*)aS)[s];
  float4 ad4 = ((const float4*)aD)[d];
  float4 c4  = ((const float4*)cst)[0];
  atomicMaxF(&m1[d * 4 + 0], lrelu(as4.x + ad4.x + w * c4.x));
  atomicMaxF(&m1[d * 4 + 1], lrelu(as4.y + ad4.y + w * c4.y));
  atomicMaxF(&m1[d * 4 + 2], lrelu(as4.z + ad4.z + w * c4.z));
  atomicMaxF(&m1[d * 4 + 3], lrelu(as4.w + ad4.w + w * c4.w));
}

__global__ void edge_sum1_kernel(const int* __restrict__ src, const int* __restrict__ dst,
                                 const float* __restrict__ ew, const float* __restrict__ lea,
                                 const float* __restrict__ aS, const float* __restrict__ aD,
                                 const float* __restrict__ cst, const float* __restrict__ m1,
                                 float* __restrict__ s1, int E, int N) {
  int i = blockIdx.x * blockDim.x + threadIdx.x;
  if (i >= E + N) return;
  int s, d; float w;
  edge_ids1(i, E, src, dst, ew, lea, s, d, w);
  float4 as4 = ((const float4*)aS)[s];
  float4 ad4 = ((const float4*)aD)[d];
  float4 c4  = ((const float4*)cst)[0];
  float4 mm  = ((const float4*)m1)[d];
  atomicAddF(&s1[d * 4 + 0], __expf(lrelu(as4.x + ad4.x + w * c4.x) - mm.x));
  atomicAddF(&s1[d * 4 + 1], __expf(lrelu(as4.y + ad4.y + w * c4.y) - mm.y));
  atomicAddF(&s1[d * 4 + 2], __expf(lrelu(as4.z + ad4.z + w * c4.z) - mm.z));
  atomicAddF(&s1[d * 4 + 3], __expf(lrelu(as4.w + ad4.w + w * c4.w) - mm.w));
}

// one thread per (edge, head): 32-channel scatter-accumulate (b128 gathers)
__global__ void edge_agg1_kernel(const int* __restrict__ src, const int* __restrict__ dst,
                                 const float* __restrict__ ew, const float* __restrict__ lea,
                                 const float* __restrict__ aS, const float* __restrict__ aD,
                                 const float* __restrict__ cst, const float* __restrict__ m1,
                                 const float* __restrict__ s1, const float* __restrict__ h1,
                                 float* __restrict__ f2, int E, int N) {
  int i = blockIdx.x * blockDim.x + threadIdx.x;
  if (i >= (E + N) * 4) return;
  int t = i >> 2, h = i & 3;
  int s, d; float w;
  edge_ids1(t, E, src, dst, ew, lea, s, d, w);

  const float4* hp = (const float4*)(h1 + (size_t)s * 128 + h * 32);
  __builtin_prefetch(hp, 0, 0);   // global_prefetch_b8: overlap gather with alpha math

  float lg    = lrelu(aS[s * 4 + h] + aD[d * 4 + h] + w * cst[h]);
  float p     = __expf(lg - m1[d * 4 + h]);
  float alpha = p / (s1[d * 4 + h] + 1e-16f);

  float* op = f2 + (size_t)d * 128 + h * 32;
#pragma unroll
  for (int c = 0; c < 8; ++c) {
    float4 v = hp[c];
    atomicAddF(&op[c * 4 + 0], alpha * v.x);
    atomicAddF(&op[c * 4 + 1], alpha * v.y);
    atomicAddF(&op[c * 4 + 2], alpha * v.z);
    atomicAddF(&op[c * 4 + 3], alpha * v.w);
  }
}

// ---------- ELU (+ b1) in place (vectorized) ----------

__global__ void elu_bias_kernel(float* __restrict__ f2, const float* __restrict__ b1, int N) {
  int i = blockIdx.x * blockDim.x + threadIdx.x;   // over N*32 float4s
  if (i >= N * 32) return;
  float4 v = ((float4*)f2)[i];
  float4 b = ((const float4*)b1)[i & 31];
  v.x += b.x; v.y += b.y; v.z += b.z; v.w += b.w;
  v.x = v.x > 0.f ? v.x : expm1f(v.x);
  v.y = v.y > 0.f ? v.y : expm1f(v.y);
  v.z = v.z > 0.f ? v.z : expm1f(v.z);
  v.w = v.w > 0.f ? v.w : expm1f(v.w);
  ((float4*)f2)[i] = v;
}

// ---------- layer-2 GEMV: h2[n] = f2[n,:] . W2 (one wave32 per node, one b128 per lane) ----------

__global__ void __launch_bounds__(128)
h2_kernel(const float* __restrict__ f2, const float* __restrict__ W2,
          float* __restrict__ h2, int N) {
  int wid  = (blockIdx.x * blockDim.x + threadIdx.x) >> 5;
  int lane = threadIdx.x & 31;
  if (wid >= N) return;
  float4 v = ((const float4*)(f2 + (size_t)wid * 128))[lane];
  float4 w = ((const float4*)W2)[lane];
  float acc = v.x * w.x + v.y * w.y + v.z * w.z + v.w * w.w;
#pragma unroll
  for (int off = 16; off; off >>= 1) acc += __shfl_xor(acc, off, 32);
  if (lane == 0) h2[wid] = acc;
}

// ---------- layer-2 edge softmax (1 head, 1 channel) ----------

__global__ void edge_max2_kernel(const int* __restrict__ src, const int* __restrict__ dst,
                                 const float* __restrict__ ew, const float* __restrict__ lea,
                                 const float* __restrict__ h2,
                                 const float* __restrict__ attS2, const float* __restrict__ attD2,
                                 const float* __restrict__ cst, float* __restrict__ m2,
                                 int E, int N) {
  int i = blockIdx.x * blockDim.x + threadIdx.x;
  if (i >= E + N) return;
  int s, d; float w;
  edge_ids1(i, E, src, dst, ew, lea, s, d, w);
  float lg = lrelu(h2[s] * attS2[0] + h2[d] * attD2[0] + w * cst[4]);
  atomicMaxF(&m2[d], lg);
}

__global__ void edge_sum2_kernel(const int* __restrict__ src, const int* __restrict__ dst,
                                 const float* __restrict__ ew, const float* __restrict__ lea,
                                 const float* __restrict__ h2,
                                 const float* __restrict__ attS2, const float* __restrict__ attD2,
                                 const float* __restrict__ cst, const float* __restrict__ m2,
                                 float* __restrict__ s2, int E, int N) {
  int i = blockIdx.x * blockDim.x + threadIdx.x;
  if (i >= E + N) return;
  int s, d; float w;
  edge_ids1(i, E, src, dst, ew, lea, s, d, w);
  float lg = lrelu(h2[s] * attS2[0] + h2[d] * attD2[0] + w * cst[4]);
  atomicAddF(&s2[d], __expf(lg - m2[d]));
}

__global__ void edge_agg2_kernel(const int* __restrict__ src, const int* __restrict__ dst,
                                 const float* __restrict__ ew, const float* __restrict__ lea,
                                 const float* __restrict__ h2,
                                 const float* __restrict__ attS2, const float* __restrict__ attD2,
                                 const float* __restrict__ cst, const float* __restrict__ m2,
                                 const float* __restrict__ s2, float* __restrict__ o2,
                                 int E, int N) {
  int i = blockIdx.x * blockDim.x + threadIdx.x;
  if (i >= E + N) return;
  int s, d; float w;
  edge_ids1(i, E, src, dst, ew, lea, s, d, w);
  float lg    = lrelu(h2[s] * attS2[0] + h2[d] * attD2[0] + w * cst[4]);
  float p     = __expf(lg - m2[d]);
  float alpha = p / (s2[d] + 1e-16f);
  atomicAddF(&o2[d], alpha * h2[s]);
}

__global__ void final_kernel(const float* __restrict__ o2, const float* __restrict__ b2,
                             float* __restrict__ out, int N) {
  int n = blockIdx.x * blockDim.x + threadIdx.x;
  if (n < N) out[n] = o2[n] + b2[0];
}

// ---------- launcher ----------

extern "C" void kernel_launch(void* const* d_in, const int* in_sizes, int n_in,
                              void* d_out, int out_size, void* d_ws, size_t ws_size,
                              hipStream_t stream) {
  const float* x     = (const float*)d_in[0];
  const int*   eidx  = (const int*)d_in[1];
  const float* ew    = (const float*)d_in[2];
  const float* W1    = (const float*)d_in[3];
  const float* attS1 = (const float*)d_in[4];
  const float* attD1 = (const float*)d_in[5];
  const float* We1   = (const float*)d_in[6];
  const float* attE1 = (const float*)d_in[7];
  const float* b1    = (const float*)d_in[8];
  const float* W2    = (const float*)d_in[9];
  const float* attS2 = (const float*)d_in[10];
  const float* attD2 = (const float*)d_in[11];
  const float* We2   = (const float*)d_in[12];
  const float* attE2 = (const float*)d_in[13];
  const float* b2    = (const float*)d_in[14];
  float* out = (float*)d_out;

  const int N = in_sizes[0] / 2;
  const int E = in_sizes[2];
  const int* src = eidx;
  const int* dst = eidx + E;

  float* ws = (float*)d_ws;
  size_t off = 0;
  float* deg  = ws + off; off += (size_t)N;
  float* wsum = ws + off; off += (size_t)N;
  float* lea  = ws + off; off += (size_t)N;
  float* h1   = ws + off; off += (size_t)N * 128;
  float* aS1  = ws + off; off += (size_t)N * 4;
  float* aD1  = ws + off; off += (size_t)N * 4;
  float* m1   = ws + off; off += (size_t)N * 4;
  float* s1   = ws + off; off += (size_t)N * 4;
  float* f2   = ws + off; off += (size_t)N * 128;
  float* h2   = ws + off; off += (size_t)N;
  float* m2   = ws + off; off += (size_t)N;
  float* s2   = ws + off; off += (size_t)N;  // s2 and o2 contiguous: zeroed together
  float* o2   = ws + off; off += (size_t)N;
  float* cst  = ws + off; off += 8;
  (void)ws_size; (void)n_in; (void)out_size;

  const int B = 256;
  auto G  = [](long n, int b) { return (int)((n + b - 1) / b); };
  auto G4 = [](long n, int b) { long t = (n + 3) / 4; return (int)((t + b - 1) / b); };

  // init (deg+wsum contiguous; s2+o2 contiguous)
  fill_kernel<<<G4(2L * N, B), B, 0, stream>>>(deg, 0.f, 2 * N);
  fill_kernel<<<G4(4L * N, B), B, 0, stream>>>(m1, -INFINITY, 4 * N);
  fill_kernel<<<G4(4L * N, B), B, 0, stream>>>(s1, 0.f, 4 * N);
  fill_kernel<<<G4(128L * N, B), B, 0, stream>>>(f2, 0.f, N * 128);
  fill_kernel<<<G4(N, B), B, 0, stream>>>(m2, -INFINITY, N);
  fill_kernel<<<G4(2L * N, B), B, 0, stream>>>(s2, 0.f, 2 * N);

  // graph stats + constants
  deg_kernel<<<G(E, B), B, 0, stream>>>(dst, ew, deg, wsum, E);
  loopea_kernel<<<G(N, B), B, 0, stream>>>(deg, wsum, lea, N);
  const_kernel<<<1, 32, 0, stream>>>(We1, attE1, We2, attE2, cst);

  // layer 1
  int tiles = (N + 15) / 16;
  gemm1_wmma_kernel<<<(tiles + 3) / 4, 128, 0, stream>>>(x, W1, h1, N);
  node_att1_kernel<<<G(4L * N, B), B, 0, stream>>>(h1, attS1, attD1, aS1, aD1, N);
  edge_max1_kernel<<<G((long)E + N, B), B, 0, stream>>>(src, dst, ew, lea, aS1, aD1, cst, m1, E, N);
  edge_sum1_kernel<<<G((long)E + N, B), B, 0, stream>>>(src, dst, ew, lea, aS1, aD1, cst, m1, s1, E, N);
  edge_agg1_kernel<<<G(4L * (E + N), B), B, 0, stream>>>(src, dst, ew, lea, aS1, aD1, cst, m1, s1,
                                                         h1, f2, E, N);
  elu_bias_kernel<<<G(32L * N, B), B, 0, stream>>>(f2, b1, N);

  // layer 2
  h2_kernel<<<(N + 3) / 4, 128, 0, stream>>>(f2, W2, h2, N);
  edge_max2_kernel<<<G((long)E + N, B), B, 0, stream>>>(src, dst, ew, lea, h2, attS2, attD2, cst, m2, E, N);
  edge_sum2_kernel<<<G((long)E + N, B), B, 0, stream>>>(src, dst, ew, lea, h2, attS2, attD2, cst, m2, s2, E, N);
  edge_agg2_kernel<<<G((long)E + N, B), B, 0, stream>>>(src, dst, ew, lea, h2, attS2, attD2, cst, m2, s2, o2, E, N);
  final_kernel<<<G(N, B), B, 0, stream>>>(o2, b2, out, N);
}